// Model1_43026982371969
// MI455X (gfx1250) — compile-verified
//
#include <hip/hip_runtime.h>
#include <math.h>

#define G_ 5000
#define T_ 64
#define NW_ 7
#define LUT_N 1024
#define NB_BLOCKS 2048
#define HALF_LOG_2PI 0.9189385332046727

typedef __attribute__((ext_vector_type(2))) float v2f;
typedef __attribute__((ext_vector_type(8))) float v8f;

// ---------------------------------------------------------------------------
// Setup: scalar constants derived from (beta, log_phi) + lgamma lookup tables.
// ---------------------------------------------------------------------------
__global__ void setup_kernel(const float* __restrict__ beta_p,
                             const float* __restrict__ log_phi_p,
                             float* __restrict__ consts,
                             float* __restrict__ lutA,
                             float* __restrict__ lutB) {
  int i = blockIdx.x * blockDim.x + threadIdx.x;
  float lp  = log_phi_p[0];
  float phi = expf(lp);
  if (i == 0) {
    consts[0] = phi;
    consts[1] = lp;
    consts[2] = lgammaf(phi);
    consts[3] = phi * lp;
    consts[4] = beta_p[0];
  }
  if (i < LUT_N) {
    float yf = (float)i;
    lutA[i] = lgammaf(yf + phi);   // lgamma(y + phi)
    lutB[i] = lgammaf(yf + 1.0f);  // lgamma(y + 1)
  }
}

// ---------------------------------------------------------------------------
// Random-walk states as a triangular matmul on the WMMA pipes:
//   states[g, t] = init[g] + sum_{s<=t} z[g,s] * exp(log_scale[g])
//                = init[g] + (z .* scale) @ U,   U[s,t] = (s <= t)
// One wave per 16x16 tile of the 5000x64 state matrix via
// V_WMMA_F32_16X16X4_F32. Branchless inner loop: padded rows (g >= G) load
// from a clamped valid address and are multiplied by scale==0, so no exec
// divergence and no per-load waits serialize the WMMA chain.
// ---------------------------------------------------------------------------
__global__ void __launch_bounds__(32) states_wmma_kernel(
    const float* __restrict__ z,
    const float* __restrict__ log_scale,
    const float* __restrict__ initv,
    float* __restrict__ iym_out) {
  const int unit     = blockIdx.x;
  const int row_tile = unit >> 2;   // group tile: 0..312
  const int col_tile = unit & 3;    // time tile:  0..3
  const int lane     = threadIdx.x; // 0..31
  const int half     = lane >> 4;   // lane-half carries the high K / high M bit
  const int l        = lane & 15;

  const int g_base = row_tile * 16;
  const int n0     = col_tile * 16;

  // A-matrix row owned by this lane: M = l, K = vgpr + 2*half (ISA 16x4 f32 layout)
  const int  gA_raw = g_base + l;
  const bool validA = (gA_raw < G_);
  const int  gA     = validA ? gA_raw : (G_ - 1);        // clamped, always legal
  const float scaleA = validA ? expf(log_scale[gA]) : 0.0f;
  const float* __restrict__ zrow = z + gA * T_ + 2 * half;

  const int nl = n0 + l;                                  // this lane's N column
  v8f acc = {};
  // Only K-chunks with kc <= n0+15 touch the lower triangle of U.
  for (int kc = 0; kc <= n0 + 12; kc += 4) {
    v2f a;
    a.x = zrow[kc]     * scaleA;                          // VGPR0: K = kc + 2h
    a.y = zrow[kc + 1] * scaleA;                          // VGPR1: K = kc + 2h + 1
    const int kA = kc + 2 * half;
    v2f b;                                                // B[k][n] = (k <= n)
    b.x = (kA     <= nl) ? 1.0f : 0.0f;
    b.y = (kA + 1 <= nl) ? 1.0f : 0.0f;
    acc = __builtin_amdgcn_wmma_f32_16x16x4_f32(
        /*neg_a=*/false, a, /*neg_b=*/false, b,
        /*c_mod=*/(short)0, acc, /*reuse_a=*/false, /*reuse_b=*/false);
  }

  // D layout: VGPR r holds (M = r + 8*half, N = l)
  for (int r = 0; r < 8; ++r) {
    int g = g_base + r + 8 * half;
    if (g < G_) iym_out[g * T_ + n0 + l] = acc[r] + initv[g];
  }
}

// ---------------------------------------------------------------------------
// Streaming NB2 log-pmf reduction: 16 B/row over N rows, b128-vectorized.
// lgamma LUTs + wday staged in LDS; iym table (1.25 MB) stays L2-resident.
// ---------------------------------------------------------------------------
__device__ __forceinline__ float nb2_term(float pr, int y, int ii, int wi,
                                          float beta, float phi, float lp,
                                          float lg_phi, float phi_lp,
                                          const float* __restrict__ iym,
                                          const float* sA, const float* sB,
                                          const float* swd) {
  const float eta = fmaf(beta, pr, iym[ii] + swd[wi]);
  const float yf  = (float)y;
  float ga, gb;
  if ((unsigned)y < (unsigned)LUT_N) {        // hot path: LDS LUT
    ga = sA[y];
    gb = sB[y];
  } else {                                     // general-correctness fallback
    ga = lgammaf(yf + phi);
    gb = lgammaf(yf + 1.0f);
  }
  // stable logaddexp(eta, log_phi)
  const float m   = fmaxf(eta, lp);
  const float lse = m + log1pf(expf(-fabsf(eta - lp)));
  return ga - lg_phi - gb + phi_lp + fmaf(yf, eta, -(yf + phi) * lse);
}

__global__ void __launch_bounds__(256) nb2_kernel(
    const float* __restrict__ price,
    const int*   __restrict__ sales,
    const int*   __restrict__ iym_idx,
    const int*   __restrict__ wday_idx,
    const float* __restrict__ wday,
    const float* __restrict__ iym,
    const float* __restrict__ consts,
    const float* __restrict__ lutA,
    const float* __restrict__ lutB,
    float* __restrict__ block_part,
    int n) {
  __shared__ float sA[LUT_N];
  __shared__ float sB[LUT_N];
  __shared__ float swd[8];
  __shared__ float sdata[256];

  const int tid = threadIdx.x;
  for (int i = tid; i < LUT_N; i += 256) {
    sA[i] = lutA[i];
    sB[i] = lutB[i];
  }
  if (tid < NW_) swd[tid] = wday[tid];
  __syncthreads();

  const float phi    = consts[0];
  const float lp     = consts[1];
  const float lg_phi = consts[2];
  const float phi_lp = consts[3];
  const float beta   = consts[4];

  const int gtid   = blockIdx.x * blockDim.x + tid;
  const int stride = gridDim.x * blockDim.x;
  const int nv     = n >> 2;                   // # of float4 groups

  const float4* __restrict__ price4 = (const float4*)price;
  const int4*   __restrict__ sales4 = (const int4*)sales;
  const int4*   __restrict__ iidx4  = (const int4*)iym_idx;
  const int4*   __restrict__ widx4  = (const int4*)wday_idx;

  float acc = 0.0f;
  for (int i = gtid; i < nv; i += stride) {
    const float4 pr = price4[i];
    const int4   y  = sales4[i];
    const int4   ii = iidx4[i];
    const int4   wi = widx4[i];
    acc += nb2_term(pr.x, y.x, ii.x, wi.x, beta, phi, lp, lg_phi, phi_lp, iym, sA, sB, swd);
    acc += nb2_term(pr.y, y.y, ii.y, wi.y, beta, phi, lp, lg_phi, phi_lp, iym, sA, sB, swd);
    acc += nb2_term(pr.z, y.z, ii.z, wi.z, beta, phi, lp, lg_phi, phi_lp, iym, sA, sB, swd);
    acc += nb2_term(pr.w, y.w, ii.w, wi.w, beta, phi, lp, lg_phi, phi_lp, iym, sA, sB, swd);
  }
  // scalar tail (n not a multiple of 4)
  for (int i = (nv << 2) + gtid; i < n; i += stride) {
    acc += nb2_term(price[i], sales[i], iym_idx[i], wday_idx[i],
                    beta, phi, lp, lg_phi, phi_lp, iym, sA, sB, swd);
  }

  sdata[tid] = acc;
  __syncthreads();
  for (int s = 128; s > 0; s >>= 1) {
    if (tid < s) sdata[tid] += sdata[tid + s];
    __syncthreads();
  }
  if (tid == 0) block_part[blockIdx.x] = sdata[0];
}

// ---------------------------------------------------------------------------
// Finalize: deterministic f64 sum of block partials + all Gaussian priors.
// ---------------------------------------------------------------------------
__global__ void __launch_bounds__(256) finalize_kernel(
    const float* __restrict__ bp, int nblocks,
    const float* __restrict__ beta_p,
    const float* __restrict__ log_phi_p,
    const float* __restrict__ z,
    const float* __restrict__ wday,
    const float* __restrict__ log_scale,
    const float* __restrict__ initv,
    float* __restrict__ out) {
  __shared__ double sd[256];
  const int tid = threadIdx.x;
  double acc = 0.0;

  for (int i = tid; i < nblocks; i += 256) acc += (double)bp[i];

  for (int i = tid; i < G_ * T_; i += 256) {
    double x = (double)z[i];
    acc += -0.5 * x * x - HALF_LOG_2PI;
  }
  for (int i = tid; i < G_; i += 256) {
    double x = (double)log_scale[i];
    acc += -0.5 * x * x - HALF_LOG_2PI;
    double y = (double)initv[i];
    acc += -0.5 * y * y - HALF_LOG_2PI;
  }
  if (tid < NW_) {
    double x = (double)wday[tid];
    acc += -0.5 * x * x - HALF_LOG_2PI;
  }
  if (tid == 0) {
    double b  = (double)beta_p[0];
    double lp = (double)log_phi_p[0];
    acc += -0.5 * b * b - HALF_LOG_2PI;
    acc += -0.5 * lp * lp - HALF_LOG_2PI;
  }

  sd[tid] = acc;
  __syncthreads();
  for (int s = 128; s > 0; s >>= 1) {
    if (tid < s) sd[tid] += sd[tid + s];
    __syncthreads();
  }
  if (tid == 0) out[0] = (float)sd[0];
}

// ---------------------------------------------------------------------------
extern "C" void kernel_launch(void* const* d_in, const int* in_sizes, int n_in,
                              void* d_out, int out_size, void* d_ws, size_t ws_size,
                              hipStream_t stream) {
  const float* beta      = (const float*)d_in[0];
  const float* log_phi   = (const float*)d_in[1];
  const float* z         = (const float*)d_in[2];
  const float* wday      = (const float*)d_in[3];
  const float* log_scale = (const float*)d_in[4];
  const float* initv     = (const float*)d_in[5];
  const float* price     = (const float*)d_in[6];
  const int*   sales     = (const int*)d_in[7];
  const int*   iym_idx   = (const int*)d_in[8];
  const int*   wday_idx  = (const int*)d_in[9];
  const int N = in_sizes[6];

  // Workspace layout (floats): [consts:8][lutA:1024][lutB:1024][iym:G*T][block_part:NB_BLOCKS]
  float* ws     = (float*)d_ws;
  float* consts = ws;
  float* lutA   = ws + 8;
  float* lutB   = ws + 8 + LUT_N;
  float* iym    = ws + 8 + 2 * LUT_N;
  float* bpart  = iym + G_ * T_;

  setup_kernel<<<(LUT_N + 255) / 256, 256, 0, stream>>>(beta, log_phi, consts, lutA, lutB);

  const int row_tiles = (G_ + 15) / 16;        // 313
  states_wmma_kernel<<<row_tiles * 4, 32, 0, stream>>>(z, log_scale, initv, iym);

  nb2_kernel<<<NB_BLOCKS, 256, 0, stream>>>(price, sales, iym_idx, wday_idx,
                                            wday, iym, consts, lutA, lutB, bpart, N);

  finalize_kernel<<<1, 256, 0, stream>>>(bpart, NB_BLOCKS, beta, log_phi, z,
                                         wday, log_scale, initv, (float*)d_out);
}